// UnifiedRankingLoss_69896297775322
// MI455X (gfx1250) — compile-verified
//
#include <hip/hip_runtime.h>

typedef float v2f __attribute__((ext_vector_type(2)));
typedef float v8f __attribute__((ext_vector_type(8)));

#define K_ITEMS 128

// ws layout: ws[0] = float loss_sum, ((uint*)ws)[1] = valid pair count
__global__ void url_init_kernel(float* ws) {
    ws[0] = 0.0f;
    ((unsigned int*)ws)[1] = 0u;
}

__global__ __launch_bounds__(256) void url_main_kernel(
    const float* __restrict__ emb_to,    // [G, K, 2]
    const float* __restrict__ emb_from,  // [G, 2]
    const int*   __restrict__ ranks,     // [G, K]
    float* __restrict__ ws) {

    __shared__ float2 sDR[K_ITEMS];   // x = distance, y = rank (float)
    __shared__ float  sPart[256];     // per-thread partial loss sums
    __shared__ float  sCnt[256];      // per-thread partial counts (exact in f32)

    const int g   = blockIdx.x;
    const int tid = threadIdx.x;

    const float fx = emb_from[g * 2 + 0];
    const float fy = emb_from[g * 2 + 1];

    // Stage 1: distances + ranks into LDS (threads 0..127, coalesced float2 loads)
    if (tid < K_ITEMS) {
        const float2 e = ((const float2*)emb_to)[g * K_ITEMS + tid];
        const float dx = e.x - fx;
        const float dy = e.y - fy;
        float2 dr;
        dr.x = sqrtf(fmaf(dx, dx, dy * dy));
        dr.y = (float)ranks[g * K_ITEMS + tid];
        sDR[tid] = dr;
    }
    __syncthreads();

    // Stage 2: each thread handles row i = tid&127, k parity = tid>>7 (64 pairs).
    // All lanes of a wave read the same sDR[k] -> conflict-free LDS broadcast.
    const int i    = tid & (K_ITEMS - 1);
    const int kpar = tid >> 7;
    const float2 me = sDR[i];
    float lsum = 0.0f;
    int   lcnt = 0;
    #pragma unroll 8
    for (int m = 0; m < K_ITEMS / 2; ++m) {
        const float2 ot = sDR[(m << 1) | kpar];
        const float dd = me.x - ot.x;          // D_i - D_k (sign exact for fp32)
        const float dr = ot.y - me.y;          // r_k - r_i (small ints, exact)
        const bool  c  = (dr > 0.0f) && (dd >= 0.0f);
        lsum += c ? dd * dr : 0.0f;
        lcnt += c ? 1 : 0;
    }
    sPart[tid] = lsum;
    sCnt[tid]  = (float)lcnt;
    __syncthreads();

    // Stage 3: wave 0 reduces 256 partials with V_WMMA_F32_16X16X4_F32.
    // With A = ones(16x4): D = A*B + C accumulates column sums of B, and the
    // total is layout-independent, so each lane may feed any 2 partials into B.
    if (tid < 32) {                   // full wave32 -> EXEC all ones (WMMA req.)
        const v2f ones = {1.0f, 1.0f};
        v8f accS = {};
        v8f accC = {};
        #pragma unroll
        for (int t = 0; t < 4; ++t) {
            v2f bs, bc;
            bs[0] = sPart[t * 64 + tid];
            bs[1] = sPart[t * 64 + 32 + tid];
            bc[0] = sCnt[t * 64 + tid];
            bc[1] = sCnt[t * 64 + 32 + tid];
            accS = __builtin_amdgcn_wmma_f32_16x16x4_f32(
                false, ones, false, bs, (short)0, accS, false, false);
            accC = __builtin_amdgcn_wmma_f32_16x16x4_f32(
                false, ones, false, bc, (short)0, accC, false, false);
        }
        // acc[0]: lane L holds column-sum N = L&15 (replicated in both halves).
        // Reduce the 16 distinct columns within each 16-lane half.
        float s = accS[0];
        float c = accC[0];
        for (int m = 1; m <= 8; m <<= 1) {
            s += __shfl_xor(s, m, 32);
            c += __shfl_xor(c, m, 32);
        }
        if (tid == 0) {
            atomicAdd(&ws[0], s);
            atomicAdd(&((unsigned int*)ws)[1], (unsigned int)(c + 0.5f));
        }
    }
}

__global__ void url_final_kernel(const float* __restrict__ ws,
                                 float* __restrict__ out) {
    const unsigned int cnt = ((const unsigned int*)ws)[1];
    out[0] = (cnt > 0u) ? (ws[0] / (float)cnt) : 0.0f;
}

extern "C" void kernel_launch(void* const* d_in, const int* in_sizes, int n_in,
                              void* d_out, int out_size, void* d_ws, size_t ws_size,
                              hipStream_t stream) {
    const float* emb_to   = (const float*)d_in[0];  // G*K*2 floats
    const float* emb_from = (const float*)d_in[1];  // G*2 floats
    const int*   ranks    = (const int*)d_in[2];    // G*K ints
    float*       out      = (float*)d_out;
    float*       ws       = (float*)d_ws;

    const int G = in_sizes[1] / 2;                  // 4096

    url_init_kernel<<<1, 1, 0, stream>>>(ws);
    url_main_kernel<<<G, 256, 0, stream>>>(emb_to, emb_from, ranks, ws);
    url_final_kernel<<<1, 1, 0, stream>>>(ws, out);
}